// GCNEncoder_48009144435526
// MI455X (gfx1250) — compile-verified
//
#include <hip/hip_runtime.h>

// ---------------------------------------------------------------------------
// GCN encoder for MI455X (gfx1250): fp32 WMMA (v_wmma_f32_16x16x4_f32) GEMMs,
// LDS-staged tiles, L2-resident float atomics for the edge scatter.
// ---------------------------------------------------------------------------

typedef __attribute__((ext_vector_type(2))) float v2f;
typedef __attribute__((ext_vector_type(8))) float v8f;

#define GCN_N_NODES 100000
#define GCN_N_EDGES 640000
#define GCN_IN_CH   128
#define GCN_HID     128
#define GCN_OUT_CH  64

// ---- degree / normalization -----------------------------------------------

__global__ void k_fill_deg(float* __restrict__ deg) {
    int i = blockIdx.x * blockDim.x + threadIdx.x;
    if (i < GCN_N_NODES) deg[i] = 1.0f;  // self-loop contributes 1
}

__global__ void k_deg_edges(const long long* __restrict__ col,
                            float* __restrict__ deg) {
    int e = blockIdx.x * blockDim.x + threadIdx.x;
    if (e < GCN_N_EDGES) atomicAdd(&deg[(int)col[e]], 1.0f);
}

__global__ void k_dinv(const float* __restrict__ deg, float* __restrict__ dinv) {
    int i = blockIdx.x * blockDim.x + threadIdx.x;
    if (i < GCN_N_NODES) {
        float d = deg[i];
        dinv[i] = (d > 0.0f) ? rsqrtf(d) : 0.0f;
    }
}

// ---- dense GEMM: H[N,NCOLS] = X[N,128] @ W[128,NCOLS]  (fp32 WMMA) --------
// One block = 16-row strip of X; wave w computes the 16x16 tile at columns
// [16w, 16w+16). K=128 marched in steps of 4 with V_WMMA_F32_16X16X4_F32.

template <int NCOLS>
__global__ void k_gemm_wmma(const float* __restrict__ X,
                            const float* __restrict__ W,
                            float* __restrict__ H) {
    constexpr int K   = 128;
    constexpr int LDW = NCOLS + 4;   // padded LDS strides -> conflict-free banks
    constexpr int LDX = K + 4;
    constexpr int NT  = (NCOLS / 16) * 32;  // threads per block

    extern __shared__ float smem[];
    float* sW = smem;                 // K * LDW
    float* sX = smem + K * LDW;       // 16 * LDX

    const int tid = threadIdx.x;

    // Stage W (coalesced)
    for (int idx = tid; idx < K * NCOLS; idx += NT) {
        int k = idx / NCOLS;
        int n = idx % NCOLS;
        sW[k * LDW + n] = W[idx];
    }
    // Stage 16x128 X tile (coalesced)
    const int row0 = blockIdx.x * 16;
    for (int idx = tid; idx < 16 * K; idx += NT) {
        int r = idx >> 7;
        int c = idx & 127;
        sX[r * LDX + c] = X[(size_t)(row0 + r) * K + c];
    }
    __syncthreads();

    const int wave  = tid >> 5;
    const int lane  = tid & 31;
    const int half  = lane >> 4;   // lane half selects K pair / row+8
    const int ln    = lane & 15;
    const int ncol0 = wave * 16;

    v8f acc = {};
#pragma unroll
    for (int k = 0; k < K; k += 4) {
        const int kk = k + (half << 1);
        v2f a, b;
        // A 16x4 f32: lanes 0-15 rows M=0..15 hold K={k,k+1}; lanes 16-31 K={k+2,k+3}
        a.x = sX[ln * LDX + kk];
        a.y = sX[ln * LDX + kk + 1];
        // B 4x16 f32: same K split across lane halves, N = ln
        b.x = sW[kk * LDW + ncol0 + ln];
        b.y = sW[(kk + 1) * LDW + ncol0 + ln];
        acc = __builtin_amdgcn_wmma_f32_16x16x4_f32(
            /*neg_a=*/false, a, /*neg_b=*/false, b,
            /*c_mod=*/(short)0, acc, /*reuse_a=*/false, /*reuse_b=*/false);
    }

    // C/D 16x16 f32 layout: VGPR j -> row j (lanes 0-15) / row j+8 (lanes 16-31)
#pragma unroll
    for (int j = 0; j < 8; ++j) {
        int r = row0 + j + half * 8;
        H[(size_t)r * NCOLS + ncol0 + ln] = acc[j];
    }
}

// ---- per-node init: out = dinv^2 * h + b  (self-loop message + bias) ------

template <int F>
__global__ void k_init_out(const float* __restrict__ h,
                           const float* __restrict__ dinv,
                           const float* __restrict__ b,
                           float* __restrict__ out) {
    int idx = blockIdx.x * blockDim.x + threadIdx.x;
    if (idx < GCN_N_NODES * F) {
        int node = idx / F;
        int c    = idx & (F - 1);
        float s  = dinv[node];
        out[idx] = s * s * h[idx] + b[c];
    }
}

// ---- edge scatter: one wave per edge, vectorized gather + float atomics ---

template <int F>
__global__ void k_edge_agg(const long long* __restrict__ row,
                           const long long* __restrict__ col,
                           const float* __restrict__ dinv,
                           const float* __restrict__ h,
                           float* __restrict__ out) {
    constexpr int VEC = F / 32;  // floats per lane (4 for F=128, 2 for F=64)
    const int lane = threadIdx.x & 31;
    const int wave = threadIdx.x >> 5;
    const int e    = blockIdx.x * (blockDim.x >> 5) + wave;
    if (e >= GCN_N_EDGES) return;

    const int r = (int)row[e];
    const int c = (int)col[e];
    const float nrm = dinv[r] * dinv[c];

    const float* hp = h   + (size_t)r * F + lane * VEC;
    float*       op = out + (size_t)c * F + lane * VEC;

    if (VEC == 4) {
        float4 v = *(const float4*)hp;
        atomicAdd(op + 0, nrm * v.x);
        atomicAdd(op + 1, nrm * v.y);
        atomicAdd(op + 2, nrm * v.z);
        atomicAdd(op + 3, nrm * v.w);
    } else {
        float2 v = *(const float2*)hp;
        atomicAdd(op + 0, nrm * v.x);
        atomicAdd(op + 1, nrm * v.y);
    }
}

__global__ void k_relu(float* __restrict__ x, int n) {
    int i = blockIdx.x * blockDim.x + threadIdx.x;
    if (i < n) x[i] = fmaxf(x[i], 0.0f);
}

// ---------------------------------------------------------------------------

extern "C" void kernel_launch(void* const* d_in, const int* in_sizes, int n_in,
                              void* d_out, int out_size, void* d_ws, size_t ws_size,
                              hipStream_t stream) {
    const float*     x   = (const float*)d_in[0];
    const long long* ei  = (const long long*)d_in[1];  // int64 [2, E]
    const float*     W1  = (const float*)d_in[2];
    const float*     b1  = (const float*)d_in[3];
    const float*     W2  = (const float*)d_in[4];
    const float*     b2  = (const float*)d_in[5];
    float*           out = (float*)d_out;

    const long long* erow = ei;               // sources
    const long long* ecol = ei + GCN_N_EDGES; // targets (aggregation index)

    // workspace layout (floats): deg | dinv | h (N*128) | a1 (N*128)
    float* ws   = (float*)d_ws;
    float* deg  = ws;
    float* dinv = ws + GCN_N_NODES;
    float* h    = ws + 2 * GCN_N_NODES;
    float* a1   = h + (size_t)GCN_N_NODES * GCN_HID;
    float* h2   = h;  // reuse h buffer for layer-2 GEMM output

    const int T = 256;

    // normalization
    k_fill_deg<<<(GCN_N_NODES + T - 1) / T, T, 0, stream>>>(deg);
    k_deg_edges<<<(GCN_N_EDGES + T - 1) / T, T, 0, stream>>>(ecol, deg);
    k_dinv<<<(GCN_N_NODES + T - 1) / T, T, 0, stream>>>(deg, dinv);

    // ---- layer 1: h = x @ W1 ; a1 = scatter(norm * h) + b1 ; relu ----
    {
        size_t smem = (size_t)(128 * (128 + 4) + 16 * 132) * sizeof(float);
        k_gemm_wmma<128><<<GCN_N_NODES / 16, 256, smem, stream>>>(x, W1, h);
    }
    k_init_out<128><<<(GCN_N_NODES * 128 + T - 1) / T, T, 0, stream>>>(h, dinv, b1, a1);
    k_edge_agg<128><<<(GCN_N_EDGES + 7) / 8, 256, 0, stream>>>(erow, ecol, dinv, h, a1);
    k_relu<<<(GCN_N_NODES * 128 + T - 1) / T, T, 0, stream>>>(a1, GCN_N_NODES * 128);

    // ---- layer 2: h2 = a1 @ W2 ; out = scatter(norm * h2) + b2 ----
    {
        size_t smem = (size_t)(128 * (64 + 4) + 16 * 132) * sizeof(float);
        k_gemm_wmma<64><<<GCN_N_NODES / 16, 128, smem, stream>>>(a1, W2, h2);
    }
    k_init_out<64><<<(GCN_N_NODES * 64 + T - 1) / T, T, 0, stream>>>(h2, dinv, b2, out);
    k_edge_agg<64><<<(GCN_N_EDGES + 7) / 8, 256, 0, stream>>>(erow, ecol, dinv, h2, out);
}